// kernel_generated_3_21887153341344
// MI455X (gfx1250) — compile-verified
//
#include <hip/hip_runtime.h>
#include <hip/hip_bf16.h>

// GEMM view of the reference:
//   y[i, s] = sum_t A[s, t] * W[i, t]
//   s = o*14+n (196 spatial), t = j*6 + l*3 + k (K = 768, == W's flat layout)
//   A[s, t] = x[j, (o-1-l) mod 14, n+k-1]  (zero outside width 0..13)
//
// One kernel, 13 blocks x 512 threads (16 waves). Block stages its 16x768
// fp32 A-tile in LDS (im2col), each wave computes one 16x16 output tile with
// V_WMMA_F32_16X16X4_F32 over 192 K-steps, B fragments read directly from W
// (contiguous per channel row, L2-resident).

typedef __attribute__((ext_vector_type(2))) float v2f;
typedef __attribute__((ext_vector_type(8))) float v8f;

#define KTOT 768
#define LDS_STRIDE 772   // 772 % 64 == 4 -> conflict-free column reads
#define MT 16
#define NSPAT 196

__global__ __launch_bounds__(512) void conv_shift_wmma_f32(
    const float* __restrict__ x,     // (1,128,14,14)
    const float* __restrict__ Wm,    // (256,128,2,3) == (256, 768)
    float* __restrict__ out)         // (1,256,14,14)
{
    __shared__ float As[MT * LDS_STRIDE];

    const int mtile = blockIdx.x;                 // 0..12
    const int lane  = threadIdx.x & 31;
    const int wv    = threadIdx.x >> 5;           // wave id 0..15 -> N tile
    const int chan  = (wv << 4) + (lane & 15);    // output channel this lane owns in B/D

    // Warm this wave's 16 W rows into cache while we build the A tile.
    __builtin_prefetch(Wm + chan * KTOT, 0, 3);

    // ---- im2col: A[m][t] = x[j, (o-1-l) mod 14, n+k-1] (width zero-padded) ----
    for (int idx = threadIdx.x; idx < MT * KTOT; idx += 512) {
        const int m = idx / KTOT;
        const int t = idx - m * KTOT;
        const int s = mtile * MT + m;
        float v = 0.0f;
        if (s < NSPAT) {
            const int o = s / 14;
            const int n = s - o * 14;
            const int j = t / 6;
            const int r = t - j * 6;
            const int l = r / 3;
            const int k = r - l * 3;
            int row = o - 1 - l;                  // in [-2, 12]
            if (row < 0) row += 14;               // circular roll of the H axis
            const int col = n + k - 1;            // width pad-1 window
            if (col >= 0 && col < 14)
                v = x[j * NSPAT + row * 14 + col];
        }
        As[m * LDS_STRIDE + t] = v;
    }
    __syncthreads();

    // ---- WMMA fragment addressing (f32 16x16x4, ISA 7.12.2 layouts) ----
    // A 16x4:  lane holds M = lane&15; VGPR0/1 = K {0,1} (lanes 0-15) or {2,3} (16-31)
    // B 4x16:  lane holds N = lane&15; same K split across the half-waves
    const int m     = lane & 15;
    const int khalf = (lane >> 4) << 1;           // 0 or 2
    const float* wrow = Wm + chan * KTOT;

    v8f acc = {};
    #pragma unroll 8
    for (int k0 = 0; k0 < KTOT; k0 += 4) {
        v2f a = *(const v2f*)&As[m * LDS_STRIDE + k0 + khalf];   // 8B aligned ds_load
        v2f b = *(const v2f*)&wrow[k0 + khalf];                  // 8B aligned global
        acc = __builtin_amdgcn_wmma_f32_16x16x4_f32(
            /*neg_a=*/false, a, /*neg_b=*/false, b,
            /*c_mod=*/(short)0, acc, /*reuse_a=*/false, /*reuse_b=*/false);
    }

    // ---- D layout: VGPR v holds (M = v + 8*(lane>=16), N = lane&15) ----
    const int mofs = (lane >> 4) << 3;            // 0 or 8
    #pragma unroll
    for (int v = 0; v < 8; ++v) {
        const int s = mtile * MT + v + mofs;
        if (s < NSPAT)
            out[chan * NSPAT + s] = acc[v];
    }
}

extern "C" void kernel_launch(void* const* d_in, const int* in_sizes, int n_in,
                              void* d_out, int out_size, void* d_ws, size_t ws_size,
                              hipStream_t stream) {
    const float* x  = (const float*)d_in[0];   // 128*14*14 = 25088 f32
    const float* Wm = (const float*)d_in[1];   // 256*128*2*3 = 196608 f32
    float* out = (float*)d_out;                // 256*14*14 = 50176 f32

    dim3 grid(13);       // ceil(196 / 16) M tiles
    dim3 block(512);     // 16 waves, one 16-channel N tile each
    conv_shift_wmma_f32<<<grid, block, 0, stream>>>(x, Wm, out);
}